// WaveletBase_43026982371459
// MI455X (gfx1250) — compile-verified
//
#include <hip/hip_runtime.h>

#define B_   64
#define C_   16
#define N_   8192
#define ROWS (B_ * C_)   // 1024 rows of 8192 floats

typedef __attribute__((ext_vector_type(4))) float v4f;

// ---------------------------------------------------------------------------
// Kernel A: per-row sums over 1024-element blocks (8 sums per row).
// The row (32 KB) is staged into LDS with CDNA5 async global->LDS b128 loads
// (ASYNCcnt), then reduced. One wave32 per 1024-block.
// ---------------------------------------------------------------------------
__global__ __launch_bounds__(256) void wave_blocksum(const float* __restrict__ x,
                                                     float* __restrict__ S) {
  __shared__ __align__(16) float lds[N_];
  const int tid = threadIdx.x;
  const int row = blockIdx.x;
  const float* gbase = x + (size_t)row * N_;   // uniform -> SGPR pair

  const unsigned ldsbase = (unsigned)(uintptr_t)(void*)lds;  // low 32 bits = LDS offset
  const unsigned toff = tid * 16u;                           // bytes
  #pragma unroll
  for (int i = 0; i < 8; ++i) {
    unsigned goff  = toff + (unsigned)i * 4096u;  // byte offset within row
    unsigned laddr = ldsbase + goff;
    asm volatile("global_load_async_to_lds_b128 %0, %1, %2"
                 :: "v"(laddr), "v"(goff), "s"(gbase) : "memory");
  }
  asm volatile("s_wait_asynccnt 0" ::: "memory");
  __syncthreads();

  const int k    = tid >> 5;   // 1024-block id == wave id within workgroup
  const int lane = tid & 31;
  float s = 0.f;
  #pragma unroll
  for (int i = 0; i < 32; ++i) s += lds[k * 1024 + i * 32 + lane];
  // wave32 butterfly reduction
  #pragma unroll
  for (int m = 16; m >= 1; m >>= 1) s += __shfl_xor(s, m, 32);
  if (lane == 0) S[row * 8 + k] = s;
}

// Derive all block averages of a row from its 8 block sums of 1024.
__device__ inline void row_avgs(const float* __restrict__ Srow,
                                float a10[8], float a11[4], float a12[2],
                                float* a13) {
  #pragma unroll
  for (int k = 0; k < 8; ++k) a10[k] = Srow[k] * (1.f / 1024.f);
  #pragma unroll
  for (int p = 0; p < 4; ++p) a11[p] = (Srow[2 * p] + Srow[2 * p + 1]) * (1.f / 2048.f);
  #pragma unroll
  for (int p = 0; p < 2; ++p)
    a12[p] = (Srow[4 * p] + Srow[4 * p + 1] + Srow[4 * p + 2] + Srow[4 * p + 3]) * (1.f / 4096.f);
  float t = 0.f;
  #pragma unroll
  for (int k = 0; k < 8; ++k) t += Srow[k];
  *a13 = t * (1.f / 8192.f);
}

// ---------------------------------------------------------------------------
// Kernel B: BN statistics per (j, channel). masked_inputs[j] is piecewise
// constant over equal-size blocks, so mean/var over (batch, length) equals
// mean/var over the distinct block values with equal weights.
// One workgroup per (j,c); at most 512 values to reduce.
// ---------------------------------------------------------------------------
__global__ __launch_bounds__(256) void wave_stats(const float* __restrict__ S,
                                                  const float* __restrict__ gamma,
                                                  const float* __restrict__ beta,
                                                  float* __restrict__ stats) {
  const int j   = blockIdx.x >> 4;
  const int c   = blockIdx.x & 15;
  const int tid = threadIdx.x;
  const int cnt = B_ << j;          // 64, 128, 256, 512 values

  float sum = 0.f, sq = 0.f;
  for (int idx = tid; idx < cnt; idx += 256) {
    const int b = idx >> j;
    const int m = idx & ((1 << j) - 1);
    const float* Srow = S + (size_t)(b * C_ + c) * 8;
    float a10[8], a11[4], a12[2], a13;
    row_avgs(Srow, a10, a11, a12, &a13);
    float v;
    if (j == 0)      v = a13;
    else if (j == 1) v = a12[m] - a13;
    else if (j == 2) v = a11[m] - a12[m >> 1];
    else             v = a10[m] - a11[m >> 1];
    sum += v;
    sq  += v * v;
  }

  __shared__ float rs[256], rq[256];
  rs[tid] = sum; rq[tid] = sq;
  __syncthreads();
  #pragma unroll
  for (int w = 128; w >= 1; w >>= 1) {
    if (tid < w) { rs[tid] += rs[tid + w]; rq[tid] += rq[tid + w]; }
    __syncthreads();
  }
  if (tid == 0) {
    const float inv  = 1.f / (float)cnt;
    const float mean = rs[0] * inv;
    const float var  = fmaxf(rq[0] * inv - mean * mean, 0.f);
    const float sc   = gamma[j * C_ + c] * rsqrtf(var + 1e-5f);
    stats[(j * C_ + c) * 2 + 0] = sc;
    stats[(j * C_ + c) * 2 + 1] = beta[j * C_ + c] - mean * sc;
  }
}

// ---------------------------------------------------------------------------
// Kernel C: splat-write the 256 MB output with NT 128-bit stores.
// One workgroup per (j, row); thread t's q-th float4 lands in 1024-block q.
// out[0][j][b][c][n] = BN(masked_input_j), out[1][j][b][c][n] = target_j.
// ---------------------------------------------------------------------------
__global__ __launch_bounds__(256) void wave_write(const float* __restrict__ S,
                                                  const float* __restrict__ stats,
                                                  float* __restrict__ out) {
  const int jbc = blockIdx.x;        // 4096 = 4 * 1024
  const int j   = jbc >> 10;
  const int r   = jbc & 1023;        // r = b*16 + c
  const int c   = r & 15;
  const int tid = threadIdx.x;

  const float* Srow = S + (size_t)r * 8;
  float a10[8], a11[4], a12[2], a13;
  row_avgs(Srow, a10, a11, a12, &a13);

  const float sc = stats[(j * C_ + c) * 2 + 0];
  const float sh = stats[(j * C_ + c) * 2 + 1];

  float bnv[8], tv[8];
  #pragma unroll
  for (int k = 0; k < 8; ++k) {
    float t, in;
    if (j == 0)      { t = a13;          in = a13; }
    else if (j == 1) { t = a12[k >> 2];  in = a12[k >> 2] - a13; }
    else if (j == 2) { t = a11[k >> 1];  in = a11[k >> 1] - a12[k >> 2]; }
    else             { t = a10[k];       in = a10[k] - a11[k >> 1]; }
    tv[k]  = t;
    bnv[k] = in * sc + sh;
  }

  float* o0 = out + (size_t)(j * ROWS + r) * N_;               // out[0][j][r]
  float* o1 = out + (size_t)((4 + j) * ROWS + r) * N_;         // out[1][j][r]
  #pragma unroll
  for (int q = 0; q < 8; ++q) {
    const int idx = q * 1024 + tid * 4;                        // block q
    v4f v0 = { bnv[q], bnv[q], bnv[q], bnv[q] };
    v4f v1 = { tv[q],  tv[q],  tv[q],  tv[q]  };
    __builtin_nontemporal_store(v0, (v4f*)(o0 + idx));
    __builtin_nontemporal_store(v1, (v4f*)(o1 + idx));
  }
}

// ---------------------------------------------------------------------------
extern "C" void kernel_launch(void* const* d_in, const int* in_sizes, int n_in,
                              void* d_out, int out_size, void* d_ws, size_t ws_size,
                              hipStream_t stream) {
  (void)in_sizes; (void)n_in; (void)out_size; (void)ws_size;
  const float* x     = (const float*)d_in[0];
  const float* gamma = (const float*)d_in[1];   // [4,16]
  const float* beta  = (const float*)d_in[2];   // [4,16]
  float* out = (float*)d_out;

  float* S     = (float*)d_ws;     // 1024 rows * 8 block sums = 8192 floats
  float* stats = S + ROWS * 8;     // 4*16*2 = 128 floats (scale, shift)

  wave_blocksum<<<ROWS, 256, 0, stream>>>(x, S);
  wave_stats  <<<64,   256, 0, stream>>>(S, gamma, beta, stats);
  wave_write  <<<4096, 256, 0, stream>>>(S, stats, out);
}